// GCNCommunityDetector_11261404250469
// MI455X (gfx1250) — compile-verified
//
#include <hip/hip_runtime.h>
#include <hip/hip_bf16.h>
#include <stdint.h>

typedef __attribute__((ext_vector_type(2))) float v2f;
typedef __attribute__((ext_vector_type(8))) float v8f;

#define HID    128
#define BN_EPS 1e-5f

// ---------------------------------------------------------------------------
// Degree / normalization:  deg = 1 (self loop) + sum over edges hitting dst,
// then dinv = rsqrt(deg) in place.
// ---------------------------------------------------------------------------
__global__ void deg_init_kernel(float* __restrict__ deg, int n) {
  int i = blockIdx.x * blockDim.x + threadIdx.x;
  if (i < n) deg[i] = 1.0f;   // self loop contributes 1
}

__global__ void deg_edges_kernel(const long long* __restrict__ dst,
                                 float* __restrict__ deg, int e) {
  int i = blockIdx.x * blockDim.x + threadIdx.x;
  if (i < e) atomicAdd(&deg[(int)dst[i]], 1.0f);
}

__global__ void rsqrt_kernel(float* __restrict__ deg, int n) {
  int i = blockIdx.x * blockDim.x + threadIdx.x;
  if (i < n) {
    float d = deg[i];
    deg[i] = (d > 0.0f) ? rsqrtf(d) : 0.0f;
  }
}

// ---------------------------------------------------------------------------
// H = X @ W  via V_WMMA_F32_16X16X4_F32 (wave32, 16x16 tile per wave).
// Block = 256 threads = 8 waves -> 16 rows x 128 cols per block.
// X tile staged with GLOBAL_LOAD_ASYNC_TO_LDS_B128 (ASYNCcnt path).
// LDS pad = +12 floats: row stride (K+12)*4 B is 16-aligned, and the WMMA
// A-fetch hits bank (12*m + k) % 64 -> all 32 lanes distinct banks.
// ---------------------------------------------------------------------------
template <int K>
__global__ __launch_bounds__(256) void gemm_wmma_f32(
    const float* __restrict__ X, const float* __restrict__ W,
    float* __restrict__ H, int nrows) {
  constexpr int KP = K + 12;
  __shared__ float Xs[16 * KP];

  const int rowBase = blockIdx.x * 16;

  // Async-stage 16 x K tile of X into LDS, 16 B per copy.
  for (int i = threadIdx.x; i < (16 * K) / 4; i += 256) {
    const int c4 = i * 4;
    const int r  = c4 / K;
    const int c  = c4 - r * K;
    int gr = rowBase + r;
    if (gr >= nrows) gr = nrows - 1;      // N = 100000 = 6250*16, never taken
    const uint64_t gaddr   = (uint64_t)(uintptr_t)(X + (long long)gr * K + c);
    const uint32_t ldsaddr = (uint32_t)(uintptr_t)(&Xs[r * KP + c]);
    asm volatile("global_load_async_to_lds_b128 %0, %1, off"
                 :: "v"(ldsaddr), "v"(gaddr) : "memory");
  }
  asm volatile("s_wait_asynccnt 0" ::: "memory");
  __syncthreads();

  const int lane = threadIdx.x & 31;
  const int wave = threadIdx.x >> 5;       // 0..7 -> 16-column tile
  const int m    = lane & 15;
  const int half = lane >> 4;              // 0: lanes 0-15, 1: lanes 16-31
  const int col  = wave * 16 + m;

  v8f acc = {};
#pragma unroll 8
  for (int k0 = 0; k0 < K; k0 += 4) {
    const int ka = k0 + 2 * half;
    v2f a, b;
    // A (16x4 f32): VGPR0/1 hold K = ka, ka+1 for row m of this lane-half.
    a.x = Xs[m * KP + ka];
    a.y = Xs[m * KP + ka + 1];
    // B (4x16 f32): VGPR0/1 hold rows ka, ka+1 at column col.
    b.x = W[(ka)     * HID + col];
    b.y = W[(ka + 1) * HID + col];
    acc = __builtin_amdgcn_wmma_f32_16x16x4_f32(
        /*neg_a=*/false, a, /*neg_b=*/false, b,
        /*c_mod=*/(short)0, acc, /*reuse_a=*/false, /*reuse_b=*/false);
  }

  // C/D layout: VGPR v holds row (v + 8*half), column = lane%16 within tile.
  if (rowBase + 16 <= nrows) {             // uniform fast path: no per-store exec dance
#pragma unroll
    for (int v = 0; v < 8; ++v) {
      H[(long long)(rowBase + v + 8 * half) * HID + col] = acc[v];
    }
  } else {
#pragma unroll
    for (int v = 0; v < 8; ++v) {
      int row = rowBase + v + 8 * half;
      if (row < nrows) H[(long long)row * HID + col] = acc[v];
    }
  }
}

// ---------------------------------------------------------------------------
// AGG[i][c] = H[i][c] * dinv[i]^2  (self-loop term)  + bias[c], float4-wide.
// Fully initializes AGG (overwrites harness poison) before edge atomics.
// ---------------------------------------------------------------------------
__global__ void agg_init_kernel(const float4* __restrict__ H4,
                                const float* __restrict__ dinv,
                                const float4* __restrict__ bias4,
                                float4* __restrict__ AGG4, int n) {
  long long t = (long long)blockIdx.x * blockDim.x + threadIdx.x;
  if (t >= (long long)n * (HID / 4)) return;
  const int i = (int)(t >> 5);    // / 32 float4 groups per row
  const int c = (int)(t & 31);
  const float di = dinv[i];
  const float s = di * di;
  const float4 h = H4[t];
  const float4 b = bias4[c];
  float4 o;
  o.x = h.x * s + b.x;  o.y = h.y * s + b.y;
  o.z = h.z * s + b.z;  o.w = h.w * s + b.w;
  AGG4[t] = o;
}

// ---------------------------------------------------------------------------
// Edge scatter: one wave per edge. 32 lanes x float4 = 128 features.
// Edge indices / norms are wave-uniform -> force to SGPRs (readfirstlane).
// Gather h[src] (L2-resident, 51 MB table), f32 atomic add into AGG[dst].
// ---------------------------------------------------------------------------
__global__ __launch_bounds__(256) void agg_edges_kernel(
    const float* __restrict__ H, const long long* __restrict__ src,
    const long long* __restrict__ dst, const float* __restrict__ dinv,
    float* __restrict__ AGG, int nedges) {
  int e = blockIdx.x * 8 + (threadIdx.x >> 5);
  if (e >= nedges) return;
  int s = (int)src[e];
  int d = (int)dst[e];
  s = __builtin_amdgcn_readfirstlane(s);
  d = __builtin_amdgcn_readfirstlane(d);
  const float nrm = dinv[s] * dinv[d];
  const int lane = threadIdx.x & 31;

  const float4 h4 = ((const float4*)(H + (long long)s * HID))[lane];
  float* out = AGG + (long long)d * HID + lane * 4;
  atomicAdd(out + 0, h4.x * nrm);
  atomicAdd(out + 1, h4.y * nrm);
  atomicAdd(out + 2, h4.z * nrm);
  atomicAdd(out + 3, h4.w * nrm);
}

// ---------------------------------------------------------------------------
// y = relu( (agg - rm) * rsqrt(rv + eps) * gamma + beta ), float4-wide.
// ---------------------------------------------------------------------------
__global__ void bn_relu_kernel(const float4* __restrict__ AGG4,
                               const float4* __restrict__ gamma4,
                               const float4* __restrict__ beta4,
                               const float4* __restrict__ rm4,
                               const float4* __restrict__ rv4,
                               float4* __restrict__ out4, int n) {
  long long t = (long long)blockIdx.x * blockDim.x + threadIdx.x;
  if (t >= (long long)n * (HID / 4)) return;
  const int c = (int)(t & 31);
  const float4 a  = AGG4[t];
  const float4 g  = gamma4[c];
  const float4 be = beta4[c];
  const float4 m  = rm4[c];
  const float4 v  = rv4[c];
  float4 o;
  o.x = fmaxf((a.x - m.x) * rsqrtf(v.x + BN_EPS) * g.x + be.x, 0.0f);
  o.y = fmaxf((a.y - m.y) * rsqrtf(v.y + BN_EPS) * g.y + be.y, 0.0f);
  o.z = fmaxf((a.z - m.z) * rsqrtf(v.z + BN_EPS) * g.z + be.z, 0.0f);
  o.w = fmaxf((a.w - m.w) * rsqrtf(v.w + BN_EPS) * g.w + be.w, 0.0f);
  out4[t] = o;
}

// ---------------------------------------------------------------------------
// Launch
// ---------------------------------------------------------------------------
extern "C" void kernel_launch(void* const* d_in, const int* in_sizes, int n_in,
                              void* d_out, int out_size, void* d_ws, size_t ws_size,
                              hipStream_t stream) {
  const float* x      = (const float*)d_in[0];
  const float* W1     = (const float*)d_in[1];
  const float* b1     = (const float*)d_in[2];
  const float* W2     = (const float*)d_in[3];
  const float* b2     = (const float*)d_in[4];
  const float* gamma1 = (const float*)d_in[5];
  const float* beta1  = (const float*)d_in[6];
  const float* rm1    = (const float*)d_in[7];
  const float* rv1    = (const float*)d_in[8];
  const float* gamma2 = (const float*)d_in[9];
  const float* beta2  = (const float*)d_in[10];
  const float* rm2    = (const float*)d_in[11];
  const float* rv2    = (const float*)d_in[12];
  const long long* ei = (const long long*)d_in[13];   // int64 [2, E]

  const int IN_DIM = 256;
  const int N = in_sizes[0] / IN_DIM;       // 100000
  const int E = in_sizes[13] / 2;           // 3200000
  const long long* src = ei;
  const long long* dst = ei + E;

  // Workspace layout: dinv[N] | bufA[N*128] | bufB[N*128]
  float* dinv = (float*)d_ws;
  float* bufA = dinv + ((N + 63) & ~63);
  float* bufB = bufA + (long long)N * HID;

  const int T = 256;
  const int gN    = (N + T - 1) / T;
  const int gE    = (E + T - 1) / T;
  const long long NH4 = (long long)N * (HID / 4);
  const int gNH4  = (int)((NH4 + T - 1) / T);
  const int gGemm = (N + 15) / 16;
  const int gEdge = (E + 7) / 8;            // one wave per edge, 8 waves/block

  // Normalization
  deg_init_kernel<<<gN, T, 0, stream>>>(dinv, N);
  deg_edges_kernel<<<gE, T, 0, stream>>>(dst, dinv, E);
  rsqrt_kernel<<<gN, T, 0, stream>>>(dinv, N);

  // Layer 1: H1 = x @ W1 -> aggregate -> BN+ReLU
  gemm_wmma_f32<256><<<gGemm, T, 0, stream>>>(x, W1, bufA, N);
  agg_init_kernel<<<gNH4, T, 0, stream>>>((const float4*)bufA, dinv,
                                          (const float4*)b1, (float4*)bufB, N);
  agg_edges_kernel<<<gEdge, T, 0, stream>>>(bufA, src, dst, dinv, bufB, E);
  bn_relu_kernel<<<gNH4, T, 0, stream>>>((const float4*)bufB, (const float4*)gamma1,
                                         (const float4*)beta1, (const float4*)rm1,
                                         (const float4*)rv1, (float4*)bufA, N);

  // Layer 2: H2 = act1 @ W2 -> aggregate into d_out -> BN+ReLU in place
  gemm_wmma_f32<128><<<gGemm, T, 0, stream>>>(bufA, W2, bufB, N);
  agg_init_kernel<<<gNH4, T, 0, stream>>>((const float4*)bufB, dinv,
                                          (const float4*)b2, (float4*)d_out, N);
  agg_edges_kernel<<<gEdge, T, 0, stream>>>(bufB, src, dst, dinv, (float*)d_out, E);
  bn_relu_kernel<<<gNH4, T, 0, stream>>>((const float4*)d_out, (const float4*)gamma2,
                                         (const float4*)beta2, (const float4*)rm2,
                                         (const float4*)rv2, (float4*)d_out, N);
}